// LightGCN_19722489823706
// MI455X (gfx1250) — compile-verified
//
#include <hip/hip_runtime.h>
#include <hip/hip_bf16.h>

#define NUM_USERS 100000
#define NUM_ITEMS 50000
#define N_NODES   150000
#define DIM       64
#define N_EDGES   4800000
#define NQ_USERS  1024
#define NQ_ITEMS  20000

typedef __attribute__((ext_vector_type(2))) float v2f;
typedef __attribute__((ext_vector_type(8))) float v8f;

// ---------------------------------------------------------------------------
// Init: acc = h = concat(user_emb, item_emb), vectorized float4.
// NUM_USERS*DIM = 6,400,000 is a multiple of 4, so the seam is float4-aligned.
// ---------------------------------------------------------------------------
__global__ void k_init(const float* __restrict__ ue, const float* __restrict__ ie,
                       float* __restrict__ acc, float* __restrict__ h) {
    const int total4 = (N_NODES * DIM) / 4;
    int i = blockIdx.x * blockDim.x + threadIdx.x;
    if (i >= total4) return;
    const int useam4 = (NUM_USERS * DIM) / 4;
    float4 v;
    if (i < useam4) v = ((const float4*)ue)[i];
    else            v = ((const float4*)ie)[i - useam4];
    ((float4*)acc)[i] = v;
    ((float4*)h)[i]   = v;
}

__global__ void k_zero(float* __restrict__ p, int total4) {
    int i = blockIdx.x * blockDim.x + threadIdx.x;
    if (i >= total4) return;
    ((float4*)p)[i] = make_float4(0.f, 0.f, 0.f, 0.f);
}

// ---------------------------------------------------------------------------
// SpMM scatter: one wave32 per edge, 2 dims (float2) per lane.
// Gathers h[src] (coalesced b64 loads, hot in L2: node matrix = 38.4MB),
// scales by edge weight, scatters with native f32 atomics into h_next.
// ---------------------------------------------------------------------------
__global__ void k_spmm(const int* __restrict__ esrc, const int* __restrict__ edst,
                       const float* __restrict__ eval,
                       const float* __restrict__ h, float* __restrict__ hn) {
    const int lane = threadIdx.x & 31;
    const int e = blockIdx.x * (blockDim.x >> 5) + (threadIdx.x >> 5);
    if (e >= N_EDGES) return;
    const int   s = esrc[e];
    const int   d = edst[e];
    const float w = eval[e];
    const float2 x = *(const float2*)(h + (long long)s * DIM + lane * 2);
    float* p = hn + (long long)d * DIM + lane * 2;
    __hip_atomic_fetch_add(p + 0, w * x.x, __ATOMIC_RELAXED, __HIP_MEMORY_SCOPE_AGENT);
    __hip_atomic_fetch_add(p + 1, w * x.y, __ATOMIC_RELAXED, __HIP_MEMORY_SCOPE_AGENT);
}

__global__ void k_accadd(float* __restrict__ acc, const float* __restrict__ hn, int total4) {
    int i = blockIdx.x * blockDim.x + threadIdx.x;
    if (i >= total4) return;
    float4 a = ((const float4*)acc)[i];
    float4 b = ((const float4*)hn)[i];
    a.x += b.x; a.y += b.y; a.z += b.z; a.w += b.w;
    ((float4*)acc)[i] = a;
}

// ---------------------------------------------------------------------------
// Gather queried rows, folding the /(N_LAYERS+1)=0.25 scale into each side.
// ---------------------------------------------------------------------------
__global__ void k_gather(const float* __restrict__ acc,
                         const int* __restrict__ users, const int* __restrict__ items,
                         float* __restrict__ ub, float* __restrict__ tb) {
    const long long NU = (long long)NQ_USERS * DIM;   // 65,536
    const long long NI = (long long)NQ_ITEMS * DIM;   // 1,280,000
    long long i = (long long)blockIdx.x * blockDim.x + threadIdx.x;
    if (i < NU) {
        int r = (int)(i >> 6), c = (int)(i & 63);
        ub[i] = 0.25f * acc[(long long)users[r] * DIM + c];
    } else if (i < NU + NI) {
        long long j = i - NU;
        int r = (int)(j >> 6), c = (int)(j & 63);
        tb[j] = 0.25f * acc[((long long)NUM_USERS + items[r]) * DIM + c];
    }
}

// ---------------------------------------------------------------------------
// GEMM 1024 x 20000 x 64 via V_WMMA_F32_16X16X4_F32 (exact f32), + sigmoid.
// One wave32 -> one 16x16 output tile. ISA VGPR layouts:
//   A 16x4:  lane&15 = M row,  (lane>>4)*2 = K pair offset, 2 VGPRs
//   B 4x16:  lane&15 = N col,  (lane>>4)*2 = K pair offset, 2 VGPRs
//   C/D:     VGPR r -> M = r + (lane>>4)*8, N = lane&15
// Output stream is 80MB -> nontemporal stores to keep L2 for node features.
// ---------------------------------------------------------------------------
__global__ void k_gemm_sigmoid(const float* __restrict__ U, const float* __restrict__ T,
                               float* __restrict__ out) {
    const int lane = threadIdx.x & 31;
    const int wave = threadIdx.x >> 5;
    const int mt = blockIdx.y * 16;                 // 64 M tiles
    const int nt = (blockIdx.x * 8 + wave) * 16;    // 1250 N tiles
    if (nt >= NQ_ITEMS) return;                     // wave-uniform exit; EXEC stays all-1s

    const int row   = lane & 15;          // M row for A, N col for B
    const int kpair = (lane >> 4) * 2;    // 0 or 2
    const float* arow = U + (mt + row) * DIM;
    const float* brow = T + (nt + row) * DIM;

    v8f c = {};
#pragma unroll
    for (int s = 0; s < 16; ++s) {
        const int k0 = s * 4;
        v2f a, b;
        a.x = arow[k0 + kpair + 0];
        a.y = arow[k0 + kpair + 1];
        b.x = brow[k0 + kpair + 0];
        b.y = brow[k0 + kpair + 1];
        // (neg_a, A, neg_b, B, c_mod, C, reuse_a, reuse_b)
        c = __builtin_amdgcn_wmma_f32_16x16x4_f32(false, a, false, b, (short)0, c,
                                                  false, false);
    }

    const int n  = lane & 15;
    const int mh = (lane >> 4) * 8;
#pragma unroll
    for (int r = 0; r < 8; ++r) {
        float x  = c[r];
        float sg = 1.0f / (1.0f + __expf(-x));
        __builtin_nontemporal_store(sg, out + (long long)(mt + mh + r) * NQ_ITEMS + (nt + n));
    }
}

// ---------------------------------------------------------------------------
// Host-side orchestration (graph-capture safe: stream launches only).
// Workspace layout (floats):
//   [0, 9.6M)      acc
//   [9.6M, 19.2M)  hA
//   [19.2M, 28.8M) hB
//   [28.8M, +64K)  gathered users (1024x64)
//   [.., +1.28M)   gathered items (20000x64)
// Total ~120.7 MB.
// ---------------------------------------------------------------------------
extern "C" void kernel_launch(void* const* d_in, const int* in_sizes, int n_in,
                              void* d_out, int out_size, void* d_ws, size_t ws_size,
                              hipStream_t stream) {
    const float* user_emb = (const float*)d_in[0];
    const float* item_emb = (const float*)d_in[1];
    const int*   edge_src = (const int*)d_in[2];
    const int*   edge_dst = (const int*)d_in[3];
    const float* edge_val = (const float*)d_in[4];
    const int*   users    = (const int*)d_in[5];
    const int*   items    = (const int*)d_in[6];
    float* out = (float*)d_out;

    const size_t NODE_F = (size_t)N_NODES * DIM;   // 9,600,000 floats
    float* ws  = (float*)d_ws;
    float* acc = ws;
    float* hA  = ws + NODE_F;
    float* hB  = ws + 2 * NODE_F;
    float* ub  = ws + 3 * NODE_F;
    float* tb  = ub + (size_t)NQ_USERS * DIM;

    const int total4 = (int)(NODE_F / 4);
    const int g4 = (total4 + 255) / 256;

    k_init<<<g4, 256, 0, stream>>>(user_emb, item_emb, acc, hA);

    float* cur = hA;
    float* nxt = hB;
    for (int layer = 0; layer < 3; ++layer) {
        k_zero<<<g4, 256, 0, stream>>>(nxt, total4);
        k_spmm<<<N_EDGES / 8, 256, 0, stream>>>(edge_src, edge_dst, edge_val, cur, nxt);
        k_accadd<<<g4, 256, 0, stream>>>(acc, nxt, total4);
        float* t = cur; cur = nxt; nxt = t;
    }

    {
        long long total = (long long)NQ_USERS * DIM + (long long)NQ_ITEMS * DIM;
        int g = (int)((total + 255) / 256);
        k_gather<<<g, 256, 0, stream>>>(acc, users, items, ub, tb);
    }

    {
        dim3 grid((NQ_ITEMS / 16 + 7) / 8, NQ_USERS / 16);  // (157, 64)
        k_gemm_sigmoid<<<grid, 256, 0, stream>>>(ub, tb, out);
    }
}